// NodeCentral_14405320311139
// MI455X (gfx1250) — compile-verified
//
#include <hip/hip_runtime.h>

#define N_NODES 50000
#define N_EDGES 800000
#define ATOM_DIM 128
#define BOND_DIM 16
#define HID 128
#define STEPS 4
#define A_COLS (BOND_DIM * BOND_DIM) /* 256 */

typedef __attribute__((ext_vector_type(2))) float v2f;
typedef __attribute__((ext_vector_type(8))) float v8f;

// ---------------------------------------------------------------------------
// Kernel 1: A = atoms @ kernel + bias   (M=50000, K=128, N=256), fp32 WMMA.
// One wave per 16x16 tile of A. 3125 * 16 = 50000 tiles = 50000 waves.
// ---------------------------------------------------------------------------
__global__ __launch_bounds__(256) void k_gemm_A(const float* __restrict__ atoms,
                                                const float* __restrict__ kern,
                                                const float* __restrict__ bias,
                                                float* __restrict__ A)
{
    const int wave = (blockIdx.x * blockDim.x + threadIdx.x) >> 5;
    const int lane = threadIdx.x & 31;
    const int tm = wave >> 4;      // 0..3124
    const int tn = wave & 15;      // 0..15
    const int m0 = tm * 16;
    const int n0 = tn * 16;
    const int lh = lane >> 4;      // lane half: selects K pair / M+8
    const int lm = lane & 15;

    const float* __restrict__ arow = atoms + (size_t)(m0 + lm) * ATOM_DIM;
    const int col = n0 + lm;

    v8f c = {};
    #pragma unroll
    for (int k = 0; k < ATOM_DIM; k += 4) {
        const int ka = k + 2 * lh;
        // A 16x4 f32 operand: VGPR0 = K=ka, VGPR1 = K=ka+1 (per-lane pair)
        v2f a = *(const v2f*)(arow + ka);
        // B 4x16 f32 operand: row-striped across lanes, mirrored K split
        v2f b;
        b.x = kern[(size_t)(ka + 0) * A_COLS + col];
        b.y = kern[(size_t)(ka + 1) * A_COLS + col];
        c = __builtin_amdgcn_wmma_f32_16x16x4_f32(false, a, false, b,
                                                  (short)0, c, false, false);
    }

    const float bv = bias[col];
    #pragma unroll
    for (int r = 0; r < 8; ++r) {
        const int m = m0 + r + 8 * lh;                 // C layout: M = r (+8 for hi half)
        A[(size_t)m * A_COLS + col] = c[r] + bv;
    }
}

// ---------------------------------------------------------------------------
// Kernel 2: per-edge matvec trans[e] = A[src] (16x16) . bonds[nbr] (16),
// segment-summed into agg[src] via f32 global atomics. 16 lanes per edge.
// ---------------------------------------------------------------------------
__global__ __launch_bounds__(256) void k_edges(const float* __restrict__ A,
                                               const float* __restrict__ bonds,
                                               const long long* __restrict__ pairs,
                                               float* __restrict__ agg)
{
    const int tid = blockIdx.x * blockDim.x + threadIdx.x;
    const int e = tid >> 4;        // edge index (grid sized exactly: 800000 edges)
    const int i = tid & 15;        // output row within the 16-vector

    const long long src = pairs[2 * e + 0];
    const long long nbr = pairs[2 * e + 1];

    // Pull a future edge's A row toward cache (emits global_prefetch_b8).
    const int ep = e + 4096;
    if (ep < N_EDGES) {
        __builtin_prefetch(A + (size_t)pairs[2 * ep] * A_COLS + i * BOND_DIM, 0, 0);
    }

    const float* __restrict__ ar = A + (size_t)src * A_COLS + i * BOND_DIM;
    const float* __restrict__ bn = bonds + (size_t)nbr * BOND_DIM;

    float y = 0.f;
    #pragma unroll
    for (int j = 0; j < BOND_DIM; ++j)
        y = fmaf(ar[j], bn[j], y);

    atomicAdd(&agg[(size_t)src * BOND_DIM + i], y);
}

// ---------------------------------------------------------------------------
// Kernel 3: result = relu(agg @ Wn + relu(bonds @ Wi)); write STEPS copies.
// Two fp32 WMMA chains (K=16 -> 4 steps each) per 16x16 output tile.
// 3125 * 8 = 25000 tiles = 25000 waves.
// ---------------------------------------------------------------------------
__global__ __launch_bounds__(256) void k_final(const float* __restrict__ agg,
                                               const float* __restrict__ bonds,
                                               const float* __restrict__ Wn,
                                               const float* __restrict__ Wi,
                                               float* __restrict__ out)
{
    const int wave = (blockIdx.x * blockDim.x + threadIdx.x) >> 5;
    const int lane = threadIdx.x & 31;
    const int tm = wave >> 3;      // 0..3124
    const int tn = wave & 7;       // 0..7
    const int m0 = tm * 16;
    const int n0 = tn * 16;
    const int lh = lane >> 4;
    const int lm = lane & 15;
    const int col = n0 + lm;

    const float* __restrict__ grow = agg   + (size_t)(m0 + lm) * BOND_DIM;
    const float* __restrict__ brow = bonds + (size_t)(m0 + lm) * BOND_DIM;

    v8f cn = {};
    v8f ce = {};
    #pragma unroll
    for (int k = 0; k < BOND_DIM; k += 4) {
        const int ka = k + 2 * lh;
        v2f ag = *(const v2f*)(grow + ka);
        v2f ab = *(const v2f*)(brow + ka);
        v2f bnod, binp;
        bnod.x = Wn[(size_t)(ka + 0) * HID + col];
        bnod.y = Wn[(size_t)(ka + 1) * HID + col];
        binp.x = Wi[(size_t)(ka + 0) * HID + col];
        binp.y = Wi[(size_t)(ka + 1) * HID + col];
        cn = __builtin_amdgcn_wmma_f32_16x16x4_f32(false, ag, false, bnod,
                                                   (short)0, cn, false, false);
        ce = __builtin_amdgcn_wmma_f32_16x16x4_f32(false, ab, false, binp,
                                                   (short)0, ce, false, false);
    }

    #pragma unroll
    for (int r = 0; r < 8; ++r) {
        const int m = m0 + r + 8 * lh;
        float e = ce[r];
        e = e > 0.f ? e : 0.f;                 // relu(bonds @ Wi)
        float v = cn[r] + e;
        v = v > 0.f ? v : 0.f;                 // relu(nodes + edges)
        const size_t off = (size_t)m * HID + col;
        #pragma unroll
        for (int s = 0; s < STEPS; ++s)
            out[(size_t)s * N_NODES * HID + off] = v;
    }
}

// ---------------------------------------------------------------------------
// Host-side launch. Stream-ordered; graph-capture safe (no sync allocs).
//   d_in: atoms, bonds, pairs(int64), kernel, bias, weight_node, weight_node_inp
//   A (50000x256 f32 = 12.8M floats) is staged in the upper half of d_out
//   (the copies-2..3 region), which kernel 3 overwrites last.
//   d_ws holds only the 3.2 MB segment-sum accumulator.
// ---------------------------------------------------------------------------
extern "C" void kernel_launch(void* const* d_in, const int* in_sizes, int n_in,
                              void* d_out, int out_size, void* d_ws, size_t ws_size,
                              hipStream_t stream)
{
    const float*     atoms = (const float*)d_in[0];
    const float*     bonds = (const float*)d_in[1];
    const long long* pairs = (const long long*)d_in[2];
    const float*     kern  = (const float*)d_in[3];
    const float*     bias  = (const float*)d_in[4];
    const float*     Wn    = (const float*)d_in[5];
    const float*     Wi    = (const float*)d_in[6];
    float* out = (float*)d_out;

    float* A   = out + (size_t)2 * N_NODES * HID;  // 12.8M floats scratch inside d_out
    float* agg = (float*)d_ws;                     // 50000*16 floats

    hipMemsetAsync(agg, 0, (size_t)N_NODES * BOND_DIM * sizeof(float), stream);

    // 1) A = atoms @ kernel + bias : 50000 waves -> 6250 blocks x 256
    k_gemm_A<<<6250, 256, 0, stream>>>(atoms, kern, bias, A);

    // 2) edge matvec + segment sum : 800000 edges * 16 lanes -> 50000 blocks x 256
    k_edges<<<50000, 256, 0, stream>>>(A, bonds, pairs, agg);

    // 3) fused output GEMMs + relu + 4x replicate : 25000 waves -> 3125 blocks x 256
    k_final<<<3125, 256, 0, stream>>>(agg, bonds, Wn, Wi, out);
}